// SelfAttentionHead_91182155694406
// MI455X (gfx1250) — compile-verified
//
#include <hip/hip_runtime.h>
#include <hip/hip_bf16.h>

typedef __attribute__((ext_vector_type(2))) float v2f;
typedef __attribute__((ext_vector_type(4))) float v4f;
typedef __attribute__((ext_vector_type(8))) float v8f;

#define N_EMB  1024
#define HEAD   64
#define SEQ_T  2048
#define BATCH  8
#define ROWS   (BATCH * SEQ_T)   // 16384

// D = A(16x4, f32) * B(4x16, f32) + C(16x16, f32)
// A frag: lanes 0-15 hold {K0,K1} of row M=lane; lanes 16-31 hold {K2,K3} of row M=lane-16
// B frag: lanes 0-15 hold {K0,K1} of col N=lane; lanes 16-31 hold {K2,K3} of col N=lane-16
// C/D   : lane 0-15 => N=lane, vgpr r => M=r; lanes 16-31 => N=lane-16, vgpr r => M=r+8
static __device__ __forceinline__ v8f wmma_f32(v2f a, v2f b, v8f c) {
    return __builtin_amdgcn_wmma_f32_16x16x4_f32(false, a, false, b, (short)0, c,
                                                 false, false);
}

static __device__ __forceinline__ v2f mk2(float x, float y) { v2f r; r.x = x; r.y = y; return r; }

// ---------------------------------------------------------------------------
// Kernel 1: Q/K/V projections.  y = x @ W^T, W is [HEAD, N_EMB] row-major.
// grid = (ROWS/16, 3), block = 128 (4 waves; wave w = head tile w).
// One wave per 16x16 output tile; K swept in macro-steps of 8 (b128 per lane
// per operand, two WMMAs with the same K-permutation on A and B).
// Two accumulators break the 256-deep WMMA accumulate chain.
// ---------------------------------------------------------------------------
__global__ __launch_bounds__(128)
void sah_proj_kernel(const float* __restrict__ x,
                     const float* __restrict__ Wk,
                     const float* __restrict__ Wq,
                     const float* __restrict__ Wv,
                     float* __restrict__ kb,
                     float* __restrict__ qb,
                     float* __restrict__ vb) {
    const int lane = threadIdx.x & 31;
    const int ht   = threadIdx.x >> 5;   // head tile 0..3
    const int half = lane >> 4;          // 0: frag K pos {0,1}, 1: pos {2,3}
    const int lr   = lane & 15;
    const int rt   = blockIdx.x;         // row tile 0..1023
    const int which = blockIdx.y;        // 0=K, 1=Q, 2=V

    const float* W   = (which == 0) ? Wk : (which == 1) ? Wq : Wv;
    float*       out = (which == 0) ? kb : (which == 1) ? qb : vb;

    const float* arow = x + (size_t)(rt * 16 + lr) * N_EMB + half * 4;
    const float* brow = W + (size_t)(ht * 16 + lr) * N_EMB + half * 4;

    v8f accA = {}, accB = {};
#pragma unroll 4
    for (int k0 = 0; k0 < N_EMB; k0 += 8) {
        v4f a4 = *(const v4f*)(arow + k0);
        v4f b4 = *(const v4f*)(brow + k0);
        accA = wmma_f32(mk2(a4.x, a4.y), mk2(b4.x, b4.y), accA);
        accB = wmma_f32(mk2(a4.z, a4.w), mk2(b4.z, b4.w), accB);
    }

    // store D tile: this lane owns rows m = r + half*8, col = ht*16 + lr
    float* obase = out + (size_t)(rt * 16 + half * 8) * HEAD + ht * 16 + lr;
#pragma unroll
    for (int r = 0; r < 8; ++r)
        obase[(size_t)r * HEAD] = accA[r] + accB[r];
}

// ---------------------------------------------------------------------------
// Kernel 2: fused causal flash attention over Q,K,V [B, T, H] fp32.
// grid = (T/16, B), block = 128.  All 4 waves of a block work on the SAME
// 16-row query tile; wave s handles key tiles j = s, s+4, s+8, ... with its
// own online-softmax state.  Partials are merged via LDS (log-sum-exp merge).
// Row sums of P are computed on the matrix pipe (P x ones) instead of lane
// shuffles; only the row-max reduction uses cross-lane ops.
// ---------------------------------------------------------------------------
__global__ __launch_bounds__(128)
void sah_attn_kernel(const float* __restrict__ Q,
                     const float* __restrict__ K,
                     const float* __restrict__ V,
                     float* __restrict__ O) {
    __shared__ float pstage[4][16 * 16];   // per-wave P relayout staging (4 KB)
    __shared__ float mred[4][16];          // per-wave row max
    __shared__ float lred[4][16];          // per-wave rescaled row sum
    __shared__ float ored[4][16 * 64];     // per-wave rescaled partial O (16 KB)

    const int lane = threadIdx.x & 31;
    const int wave = threadIdx.x >> 5;     // key-phase 0..3
    const int half = lane >> 4;
    const int lr   = lane & 15;
    const int b    = blockIdx.y;
    const int qt   = blockIdx.x;           // query tile 0..127
    const int q0   = qt * 16;

    const float* Qb = Q + (size_t)b * SEQ_T * HEAD;
    const float* Kb = K + (size_t)b * SEQ_T * HEAD;
    const float* Vb = V + (size_t)b * SEQ_T * HEAD;
    float* lds = pstage[wave];

    // Preload Q fragments for this query tile: 16 v2f frags (K = H = 64).
    v2f qfrag[16];
    {
        const float* qrow = Qb + (size_t)(q0 + lr) * HEAD + half * 4;
#pragma unroll
        for (int c = 0; c < 8; ++c) {
            v4f t = *(const v4f*)(qrow + c * 8);
            qfrag[2 * c]     = mk2(t.x, t.y);
            qfrag[2 * c + 1] = mk2(t.z, t.w);
        }
    }

    const v2f ones = mk2(1.0f, 1.0f);

    v8f acc0 = {}, acc1 = {}, acc2 = {}, acc3 = {};
    v8f lacc = {};                         // running row sums (replicated over lanes)
    float m_i[8];
#pragma unroll
    for (int r = 0; r < 8; ++r) m_i[r] = -__builtin_inff();

    const float scale = 0.03125f;   // N_EMB^-0.5 = 1/32

    for (int j = wave; j <= qt; j += 4) {
        const int k0r = j * 16;

        // ---- S = Q * K_tile^T  (16 WMMAs, K-dim = 64, two chains) ----
        v8f sA = {}, sB = {};
        {
            const float* krow = Kb + (size_t)(k0r + lr) * HEAD + half * 4;
#pragma unroll
            for (int c = 0; c < 8; ++c) {
                v4f t = *(const v4f*)(krow + c * 8);
                sA = wmma_f32(qfrag[2 * c],     mk2(t.x, t.y), sA);
                sB = wmma_f32(qfrag[2 * c + 1], mk2(t.z, t.w), sB);
            }
        }

        // ---- scale + causal mask + online softmax (max via shuffles) ----
        const bool diag = (j == qt);
#pragma unroll
        for (int r = 0; r < 8; ++r) {
            const int m = r + half * 8;           // query row within tile
            float sv = (sA[r] + sB[r]) * scale;
            if (diag && lr > m) sv = -__builtin_inff();

            float rm = sv;
            rm = fmaxf(rm, __shfl_xor(rm, 1, 32));
            rm = fmaxf(rm, __shfl_xor(rm, 2, 32));
            rm = fmaxf(rm, __shfl_xor(rm, 4, 32));
            rm = fmaxf(rm, __shfl_xor(rm, 8, 32));

            float mn    = fmaxf(m_i[r], rm);
            float alpha = __expf(m_i[r] - mn);    // 0 when m_i == -inf
            m_i[r] = mn;

            float pv = __expf(sv - mn);           // masked lanes -> exp(-inf) = 0

            lacc[r] *= alpha;
            acc0[r] *= alpha;
            acc1[r] *= alpha;
            acc2[r] *= alpha;
            acc3[r] *= alpha;

            lds[m * 16 + lr] = pv;                // stage P for C->A relayout
        }

        // ---- O += P * V_tile, l += P * ones  (20 WMMAs, 5 indep. chains) ----
#pragma unroll
        for (int kk = 0; kk < 4; ++kk) {
            v2f pa = *(const v2f*)&lds[lr * 16 + kk * 4 + half * 2];
            const float* vp = Vb + (size_t)(k0r + kk * 4 + half * 2) * HEAD + lr;
            acc0 = wmma_f32(pa, mk2(vp[0],  vp[HEAD +  0]), acc0);
            acc1 = wmma_f32(pa, mk2(vp[16], vp[HEAD + 16]), acc1);
            acc2 = wmma_f32(pa, mk2(vp[32], vp[HEAD + 32]), acc2);
            acc3 = wmma_f32(pa, mk2(vp[48], vp[HEAD + 48]), acc3);
            lacc = wmma_f32(pa, ones, lacc);      // row sums of P, all lanes
        }
    }

    // ---- publish per-wave row maxima ----
    if (lr == 0) {
#pragma unroll
        for (int r = 0; r < 8; ++r)
            mred[wave][r + half * 8] = m_i[r];
    }
    __syncthreads();

    // ---- rescale own partial to the global row max, publish l and O ----
#pragma unroll
    for (int r = 0; r < 8; ++r) {
        const int row = r + half * 8;
        float ms = fmaxf(fmaxf(mred[0][row], mred[1][row]),
                         fmaxf(mred[2][row], mred[3][row]));
        float al = __expf(m_i[r] - ms);           // 0 for empty partials
        acc0[r] *= al;
        acc1[r] *= al;
        acc2[r] *= al;
        acc3[r] *= al;
        if (lr == 0) lred[wave][row] = lacc[r] * al;
        float* orow = &ored[wave][row * HEAD + lr];
        orow[0]  = acc0[r];
        orow[16] = acc1[r];
        orow[32] = acc2[r];
        orow[48] = acc3[r];
    }
    __syncthreads();

    // ---- combine: wave w sums head columns [w*16, w*16+16) over 4 partials ----
    float* ob = O + ((size_t)b * SEQ_T + q0 + half * 8) * HEAD + wave * 16 + lr;
#pragma unroll
    for (int r = 0; r < 8; ++r) {
        const int row = r + half * 8;
        float lt = (lred[0][row] + lred[1][row]) + (lred[2][row] + lred[3][row]);
        const int c = row * HEAD + wave * 16 + lr;
        float s = (ored[0][c] + ored[1][c]) + (ored[2][c] + ored[3][c]);
        ob[(size_t)r * HEAD] = s / lt;
    }
}

// ---------------------------------------------------------------------------
extern "C" void kernel_launch(void* const* d_in, const int* in_sizes, int n_in,
                              void* d_out, int out_size, void* d_ws, size_t ws_size,
                              hipStream_t stream) {
    const float* x  = (const float*)d_in[0];
    const float* Wk = (const float*)d_in[1];
    const float* Wq = (const float*)d_in[2];
    const float* Wv = (const float*)d_in[3];

    float* kb = (float*)d_ws;                       // [ROWS, HEAD] = 4 MB
    float* qb = kb + (size_t)ROWS * HEAD;           // 4 MB
    float* vb = qb + (size_t)ROWS * HEAD;           // 4 MB

    sah_proj_kernel<<<dim3(ROWS / 16, 3), 128, 0, stream>>>(x, Wk, Wq, Wv, kb, qb, vb);
    sah_attn_kernel<<<dim3(SEQ_T / 16, BATCH), 128, 0, stream>>>(qb, kb, vb,
                                                                 (float*)d_out);
}